// FCOSDetect_66468913873271
// MI455X (gfx1250) — compile-verified
//
#include <hip/hip_runtime.h>
#include <stdint.h>

#define NLVL 5
#define NLOC 27280
#define NPAD 32768u
#define KTOP 1000
#define NB 32
#define NCLS 8
#define SCORE_THR 0.05f
#define IOU_THR 0.5f
#define IMG_W 1280.0f
#define IMG_H 1024.0f

typedef uint32_t u32x4 __attribute__((ext_vector_type(4)));
typedef uint32_t u32x8 __attribute__((ext_vector_type(8)));

struct Ptrs {
    const float* cls[NLVL];
    const float* reg[NLVL];
    const float* ctr[NLVL];
};

// ---------------------------------------------------------------------------
// Tensor Data Mover helpers (CDNA5 ISA ch.8): 1-D tile load, global -> LDS.
// Descriptor packed per D# group0/group1 layout; issued via inline asm so it
// is portable across the 5-arg (ROCm 7.2) and 6-arg (therock) builtin forms.
// Must be called wave-uniformly (we call from a single wave).
// ---------------------------------------------------------------------------
__device__ __forceinline__ void tdm_load_1d(uint32_t lds_off, const void* gp,
                                            uint32_t nelem, uint32_t dsz_code)
{
    unsigned long long gaddr = (unsigned long long)(uintptr_t)gp;
    u32x4 g0; u32x8 g1;
    // group0: count=1, is_restore=0, gather=0, lds_addr, global_addr[56:0], type=2
    g0[0] = 1u;
    g0[1] = lds_off;
    g0[2] = (uint32_t)(gaddr & 0xFFFFFFFFu);
    g0[3] = (uint32_t)((gaddr >> 32) & 0x1FFFFFFu) | (2u << 30);
    // group1: wg_mask=0, data_size, tensor_dim0=nelem, tensor_dim1=1,
    //         tile_dim0=nelem, tile_dim1=1, stride0=nelem
    g1[0] = (dsz_code << 16);
    g1[1] = (nelem & 0xFFFFu) << 16;
    g1[2] = (nelem >> 16) | (1u << 16);
    g1[3] = (nelem & 0xFFFFu) << 16;
    g1[4] = 1u;
    g1[5] = nelem;
    g1[6] = 0u;
    g1[7] = 0u;
    asm volatile("tensor_load_to_lds %0, %1" :: "s"(g0), "s"(g1) : "memory");
}

__device__ __forceinline__ void tdm_wait0()
{
#if __has_builtin(__builtin_amdgcn_s_wait_tensorcnt)
    __builtin_amdgcn_s_wait_tensorcnt((short)0);
#else
    asm volatile("s_wait_tensorcnt 0x0" ::: "memory");
#endif
}

// ---------------------------------------------------------------------------
// Kernel 1: fused sigmoid / max / argmax / centerness / exp-decode / key pack.
// One thread per (batch, padded location). Single streaming pass over HBM.
// ---------------------------------------------------------------------------
__global__ void fcos_prep(Ptrs P,
                          unsigned long long* __restrict__ keys,
                          int* __restrict__ labels,
                          float4* __restrict__ boxes)
{
    int g = blockIdx.x * blockDim.x + threadIdx.x;
    int b = blockIdx.y;
    if (g >= (int)NPAD) return;

    unsigned long long* keyrow = keys + (size_t)b * NPAD;
    if (g >= NLOC) { keyrow[g] = 0ull; return; }

    // Branchless-ish level select on immediates (no constant-memory loop).
    int lvl, off, W, HW; float stride;
    if      (g < 20480) { lvl = 0; off = 0;     W = 160; HW = 20480; stride = 8.f;   }
    else if (g < 25600) { lvl = 1; off = 20480; W = 80;  HW = 5120;  stride = 16.f;  }
    else if (g < 26880) { lvl = 2; off = 25600; W = 40;  HW = 1280;  stride = 32.f;  }
    else if (g < 27200) { lvl = 3; off = 26880; W = 20;  HW = 320;   stride = 64.f;  }
    else                { lvl = 4; off = 27200; W = 10;  HW = 80;    stride = 128.f; }
    int i = g - off;

    const float* cls = P.cls[lvl] + (size_t)b * NCLS * HW + i;
    const float* rg  = P.reg[lvl] + (size_t)b * 4 * HW + i;
    const float* ct  = P.ctr[lvl] + (size_t)b * HW + i;
    __builtin_prefetch(rg, 0, 0);   // global_prefetch_b8: warm reg while doing cls max

    // argmax over logits == argmax over sigmoids (monotone); strict > -> first max
    float best = cls[0]; int bi = 0;
#pragma unroll
    for (int c = 1; c < NCLS; ++c) {
        float v = cls[(size_t)c * HW];
        if (v > best) { best = v; bi = c; }
    }
    float sc    = 1.f / (1.f + expf(-best));
    float cc    = 1.f / (1.f + expf(-ct[0]));
    float score = sqrtf(sc * cc);

    float dl = expf(rg[0])              * stride;
    float dt = expf(rg[(size_t)HW])     * stride;
    float dr = expf(rg[(size_t)2 * HW]) * stride;
    float db = expf(rg[(size_t)3 * HW]) * stride;
    int   w  = i % W, h = i / W;
    float cx = ((float)w + 0.5f) * stride;
    float cy = ((float)h + 0.5f) * stride;

    labels[(size_t)b * NLOC + g] = bi + 1;
    boxes [(size_t)b * NLOC + g] = make_float4(cx - dl, cy - dt, cx + dr, cy + db);

    // score > 0 always => IEEE bits order-preserving. Low word = ~index so equal
    // scores pick the smaller index first (matches jax.lax.top_k tie-break).
    unsigned sbits = __float_as_uint(score);
    keyrow[g] = ((unsigned long long)sbits << 32) |
                (unsigned long long)(0xFFFFFFFFu - (unsigned)g);
}

// ---------------------------------------------------------------------------
// Kernel 2: per-batch exact top-1000 via full bitonic sort of 32768 u64 keys
// held entirely in LDS (256 KB of the WGP's 320 KB). Keys staged from global
// with a single TDM descriptor; synced with s_wait_tensorcnt.
// ---------------------------------------------------------------------------
__global__ void fcos_sort(const unsigned long long* __restrict__ keys,
                          const int* __restrict__ labels,
                          const float4* __restrict__ boxes,
                          float* __restrict__ top_s,
                          int* __restrict__ top_l,
                          float4* __restrict__ top_b)
{
    extern __shared__ unsigned long long smem[];
    int b   = blockIdx.x;
    int tid = threadIdx.x;
    int nt  = blockDim.x;

    if (tid == 0) {
        tdm_load_1d(0u, keys + (size_t)b * NPAD, NPAD, 3u /*8B elems*/);
        tdm_wait0();
    }
    __syncthreads();

    // Descending bitonic sort over NPAD u64 keys, 1024 threads.
    for (unsigned k = 2; k <= NPAD; k <<= 1) {
        for (unsigned j = k >> 1; j > 0; j >>= 1) {
            for (unsigned i = tid; i < NPAD; i += nt) {
                unsigned l = i ^ j;
                if (l > i) {
                    unsigned long long a = smem[i], c = smem[l];
                    bool desc = ((i & k) == 0);
                    if (desc ? (a < c) : (a > c)) { smem[i] = c; smem[l] = a; }
                }
            }
            __syncthreads();
        }
    }

    for (int r = tid; r < KTOP; r += nt) {
        unsigned long long key = smem[r];
        float    s   = __uint_as_float((unsigned)(key >> 32));
        unsigned idx = 0xFFFFFFFFu - (unsigned)(key & 0xFFFFFFFFull);
        top_s[(size_t)b * KTOP + r] = s;
        top_l[(size_t)b * KTOP + r] = labels[(size_t)b * NLOC + idx];
        top_b[(size_t)b * KTOP + r] = boxes [(size_t)b * NLOC + idx];
    }
}

// ---------------------------------------------------------------------------
// Kernel 3: class-aware greedy NMS (sequential pivots, barrier per pivot —
// exactly the fori_loop semantics) + clip + masked outputs. One WG per batch.
// Working set (top_b/top_s/top_l, 24 KB) is DMA'd into LDS with three TDM
// descriptors, then re-read 1000x from LDS during suppression.
//
// Dynamic-LDS byte layout (offsets are descriptor lds_addr immediates):
//   [    0, 16000)  sb : float4[1000]  raw boxes        (TDM)
//   [16000, 20000)  ss : float [1000]  scores           (TDM)
//   [20000, 24000)  sl : int   [1000]  labels           (TDM)
//   [24000, 40000)  so : float4[1000]  class-offset boxes
//   [40000, 44000)  sk : int   [1000]  keep flags
//   [44000, 44004)  sspan
// ---------------------------------------------------------------------------
#define L3_SB   0
#define L3_SS   16000
#define L3_SL   20000
#define L3_SO   24000
#define L3_SK   40000
#define L3_SP   44000
#define L3_BYTES 44032

__global__ void fcos_nms(const float* __restrict__ top_s,
                         const int* __restrict__ top_l,
                         const float4* __restrict__ top_b,
                         float* __restrict__ out)
{
    extern __shared__ char dsm[];
    float4* sb    = (float4*)(dsm + L3_SB);
    float*  ss    = (float*) (dsm + L3_SS);
    int*    sl    = (int*)   (dsm + L3_SL);
    float4* so    = (float4*)(dsm + L3_SO);
    int*    sk    = (int*)   (dsm + L3_SK);
    float*  sspan = (float*) (dsm + L3_SP);

    int b = blockIdx.x, tid = threadIdx.x, nt = blockDim.x;

    if (tid == 0) {
        tdm_load_1d(L3_SB, top_b + (size_t)b * KTOP, KTOP * 4, 2u /*4B elems*/);
        tdm_load_1d(L3_SS, top_s + (size_t)b * KTOP, KTOP,     2u);
        tdm_load_1d(L3_SL, top_l + (size_t)b * KTOP, KTOP,     2u);
        tdm_wait0();
    }
    __syncthreads();

    for (int r = tid; r < KTOP; r += nt)
        sk[r] = (ss[r] >= SCORE_THR) ? 1 : 0;
    __syncthreads();

    if (tid == 0) {
        float mx = -INFINITY, mn = INFINITY; int anyv = 0;
        for (int r = 0; r < KTOP; ++r) {
            if (sk[r]) {
                float4 v = sb[r];
                mx = fmaxf(mx, fmaxf(fmaxf(v.x, v.y), fmaxf(v.z, v.w)));
                mn = fminf(mn, fminf(fminf(v.x, v.y), fminf(v.z, v.w)));
                anyv = 1;
            }
        }
        *sspan = anyv ? (mx - mn + 1.0f) : 1.0f;
    }
    __syncthreads();

    float span = *sspan;
    for (int r = tid; r < KTOP; r += nt) {
        float  o = (float)sl[r] * span;
        float4 v = sb[r];
        so[r] = make_float4(v.x + o, v.y + o, v.z + o, v.w + o);
    }

    for (int i = 0; i < KTOP; ++i) {
        __syncthreads();
        if (sk[i]) {
            float4 bi = so[i];
            float  a1 = (bi.z - bi.x) * (bi.w - bi.y);
            for (int j = tid; j < KTOP; j += nt) {
                if (j > i && sk[j]) {
                    float4 bj = so[j];
                    float x1 = fmaxf(bi.x, bj.x), y1 = fmaxf(bi.y, bj.y);
                    float x2 = fminf(bi.z, bj.z), y2 = fminf(bi.w, bj.w);
                    float inter = fmaxf(x2 - x1, 0.f) * fmaxf(y2 - y1, 0.f);
                    float a2 = (bj.z - bj.x) * (bj.w - bj.y);
                    float iou = inter / (a1 + a2 - inter + 1e-9f);
                    if (iou > IOU_THR) sk[j] = 0;
                }
            }
        }
    }
    __syncthreads();

    const size_t BK = (size_t)NB * KTOP;
    float* out_s = out;
    float* out_l = out + BK;
    float* out_b = out + 2 * BK;   // [B,K,4]
    float* out_k = out + 6 * BK;

    for (int r = tid; r < KTOP; r += nt) {
        int    keep = sk[r];
        float4 v    = sb[r];
        float4 cb   = make_float4(fminf(fmaxf(v.x, 0.f), IMG_W - 1.f),
                                  fminf(fmaxf(v.y, 0.f), IMG_H - 1.f),
                                  fminf(fmaxf(v.z, 0.f), IMG_W - 1.f),
                                  fminf(fmaxf(v.w, 0.f), IMG_H - 1.f));
        size_t o1 = (size_t)b * KTOP + r;
        out_s[o1]         = keep ? ss[r] : 0.f;
        out_l[o1]         = keep ? (float)sl[r] : 0.f;
        out_b[o1 * 4 + 0] = keep ? cb.x : 0.f;
        out_b[o1 * 4 + 1] = keep ? cb.y : 0.f;
        out_b[o1 * 4 + 2] = keep ? cb.z : 0.f;
        out_b[o1 * 4 + 3] = keep ? cb.w : 0.f;
        out_k[o1]         = keep ? 1.f : 0.f;
    }
}

// ---------------------------------------------------------------------------
extern "C" void kernel_launch(void* const* d_in, const int* in_sizes, int n_in,
                              void* d_out, int out_size, void* d_ws, size_t ws_size,
                              hipStream_t stream)
{
    (void)in_sizes; (void)n_in; (void)out_size; (void)ws_size;

    Ptrs P;
    for (int l = 0; l < NLVL; ++l) {          // dict order: cls,reg,ctr per level
        P.cls[l] = (const float*)d_in[3 * l + 0];
        P.reg[l] = (const float*)d_in[3 * l + 1];
        P.ctr[l] = (const float*)d_in[3 * l + 2];
    }

    char* ws = (char*)d_ws;
    const size_t KEYS_B = (size_t)NB * NPAD * 8;        //  8,388,608
    const size_t LAB_B  = (size_t)NB * NLOC * 4;        //  3,491,840
    const size_t BOX_B  = (size_t)NB * NLOC * 16;       // 13,967,360
    const size_t TS_B   = (size_t)NB * KTOP * 4;
    const size_t TL_B   = (size_t)NB * KTOP * 4;

    unsigned long long* keys   = (unsigned long long*)(ws);
    int*                labels = (int*)   (ws + KEYS_B);
    float4*             boxes  = (float4*)(ws + KEYS_B + LAB_B);
    float*              top_s  = (float*) (ws + KEYS_B + LAB_B + BOX_B);
    int*                top_l  = (int*)   (ws + KEYS_B + LAB_B + BOX_B + TS_B);
    float4*             top_b  = (float4*)(ws + KEYS_B + LAB_B + BOX_B + TS_B + TL_B);

    dim3 g1((NPAD + 255) / 256, NB);
    fcos_prep<<<g1, dim3(256), 0, stream>>>(P, keys, labels, boxes);

    // One workgroup per batch; 256 KB dynamic LDS for the 32768 u64 keys.
    fcos_sort<<<NB, 1024, NPAD * 8, stream>>>(keys, labels, boxes,
                                              top_s, top_l, top_b);

    fcos_nms<<<NB, 1024, L3_BYTES, stream>>>(top_s, top_l, top_b, (float*)d_out);
}